// SimpleGAT_41455024341069
// MI455X (gfx1250) — compile-verified
//
#include <hip/hip_runtime.h>
#include <hip/hip_bf16.h>

typedef float v2f __attribute__((ext_vector_type(2)));
typedef float v8f __attribute__((ext_vector_type(8)));

#define IN_CH  128
#define OUT_CH 64
#define NEG_SLOPE 0.2f
// LDS layout for W: Wp[kpair*KP_STRIDE + 2*c + j] = W[2*kpair + j][c]
// KP_STRIDE = 160 floats: 160 mod 64 == 32, so the two half-waves (k-pairs
// differing by 1 row) hit disjoint 32-bank sets -> conflict-free ds_load_b64.
#define KP_STRIDE 160

// ---------------------------------------------------------------------------
// K0: zero mkey / denom and the accumulation region of d_out
// ---------------------------------------------------------------------------
__global__ __launch_bounds__(256) void gat_init(unsigned* __restrict__ mkey,
                                                float* __restrict__ denom,
                                                float* __restrict__ out,
                                                int n, int total_out) {
    int i = blockIdx.x * 256 + threadIdx.x;
    if (i < n) { mkey[i] = 0u; denom[i] = 0.0f; }
    if (i < total_out) out[i] = 0.0f;
}

// ---------------------------------------------------------------------------
// K1: h = x @ W  with V_WMMA_F32_16X16X4_F32 (full fp32; op is HBM-bound so
// fp32 matrix path costs nothing vs bf16 and matches reference arithmetic).
// One wave computes a 16-row x 64-col block; K loop steps by 4.
// A layout (ISA 7.12.2, 32-bit A 16x4): lanes 0-15 hold M=lane, K={k,k+1};
// lanes 16-31 hold M=lane-16, K={k+2,k+3}.  B mirrored (4x16): VGPR0/1 =
// rows k,k+1 in lanes 0-15 and rows k+2,k+3 in lanes 16-31 -> with the
// pair-contiguous LDS layout each B operand is one ds_load_b64.
// ---------------------------------------------------------------------------
__global__ __launch_bounds__(256) void gat_gemm_wmma(const float* __restrict__ x,
                                                     const float* __restrict__ W,
                                                     float* __restrict__ h,
                                                     int nrows) {
    __shared__ float Wp[(IN_CH / 2) * KP_STRIDE];   // 64 * 160 * 4B = 40 KB

    // stage W pair-interleaved: unit = (kpair, cpair)
    for (int u = threadIdx.x; u < (IN_CH / 2) * (OUT_CH / 2); u += 256) {
        int kp = u >> 5;            // k-pair index, 0..63
        int cp = u & 31;            // column pair, 0..31
        float2 w0 = *(const float2*)(W + (size_t)(2 * kp)     * OUT_CH + 2 * cp);
        float2 w1 = *(const float2*)(W + (size_t)(2 * kp + 1) * OUT_CH + 2 * cp);
        float4 v = make_float4(w0.x, w1.x, w0.y, w1.y);
        *(float4*)(&Wp[kp * KP_STRIDE + 4 * cp]) = v;   // ds_store_b128
    }
    __syncthreads();

    const int wave    = threadIdx.x >> 5;
    const int lane    = threadIdx.x & 31;
    const int waveId  = blockIdx.x * 8 + wave;
    const int rowBase = waveId * 16;
    if (rowBase >= nrows) return;

    const int lo = lane & 15;   // column within 16-wide tile / A row
    const int hi = lane >> 4;   // which K pair of the 4-wide K step

    v8f acc0 = {}, acc1 = {}, acc2 = {}, acc3 = {};
    const float* xrow = x + (size_t)(rowBase + lo) * IN_CH;

    for (int k = 0; k < IN_CH; k += 4) {
        const int ka = k + 2 * hi;                       // even
        v2f a = *(const v2f*)(xrow + ka);                // global_load_b64
        const float* bbase = Wp + (ka >> 1) * KP_STRIDE + 2 * lo;
#pragma unroll
        for (int t = 0; t < 4; ++t) {
            v2f b = *(const v2f*)(bbase + 32 * t);       // ds_load_b64, pair-contiguous
            v8f* accp = (t == 0) ? &acc0 : (t == 1) ? &acc1 : (t == 2) ? &acc2 : &acc3;
            *accp = __builtin_amdgcn_wmma_f32_16x16x4_f32(
                /*neg_a=*/false, a, /*neg_b=*/false, b,
                /*c_mod=*/(short)0, *accp, /*reuse_a=*/false, /*reuse_b=*/false);
        }
    }

    // C/D layout: VGPR r holds row rowBase + r (+8 for upper half-wave), col = t*16 + lo
#pragma unroll
    for (int t = 0; t < 4; ++t) {
        v8f acc = (t == 0) ? acc0 : (t == 1) ? acc1 : (t == 2) ? acc2 : acc3;
#pragma unroll
        for (int r = 0; r < 8; ++r) {
            int row = rowBase + r + 8 * hi;
            h[(size_t)row * OUT_CH + t * 16 + lo] = acc[r];
        }
    }
}

// ---------------------------------------------------------------------------
// K2: s_src[n] = h[n,:] . a_src ; s_dst[n] = h[n,:] . a_dst  (one wave per node)
// ---------------------------------------------------------------------------
__global__ __launch_bounds__(256) void gat_scores(const float* __restrict__ h,
                                                  const float* __restrict__ a_src,
                                                  const float* __restrict__ a_dst,
                                                  float* __restrict__ s_src,
                                                  float* __restrict__ s_dst, int n) {
    int node = (blockIdx.x * 256 + threadIdx.x) >> 5;
    int lane = threadIdx.x & 31;
    if (node >= n) return;
    float2 hv = ((const float2*)(h + (size_t)node * OUT_CH))[lane];
    float2 as = ((const float2*)a_src)[lane];
    float2 ad = ((const float2*)a_dst)[lane];
    float ps = hv.x * as.x + hv.y * as.y;
    float pd = hv.x * ad.x + hv.y * ad.y;
#pragma unroll
    for (int off = 16; off > 0; off >>= 1) {
        ps += __shfl_xor(ps, off, 32);
        pd += __shfl_xor(pd, off, 32);
    }
    if (lane == 0) { s_src[node] = ps; s_dst[node] = pd; }
}

// monotone float->uint order key (for atomic segment max)
__device__ __forceinline__ unsigned f32_key(float f) {
    int i = __float_as_int(f);
    return (unsigned)(i ^ ((i >> 31) | 0x80000000));
}
__device__ __forceinline__ float key_f32(unsigned k) {
    int i = (k & 0x80000000u) ? (int)(k ^ 0x80000000u) : (int)~k;
    return __int_as_float(i);
}

// ---------------------------------------------------------------------------
// K3: per-edge leaky-relu logits + segment max over dst (self-loops appended)
// ---------------------------------------------------------------------------
__global__ __launch_bounds__(256) void gat_logits(const int* __restrict__ ei,
                                                  const float* __restrict__ s_src,
                                                  const float* __restrict__ s_dst,
                                                  float* __restrict__ logits,
                                                  unsigned* __restrict__ mkey,
                                                  int E, int EN) {
    int e = blockIdx.x * 256 + threadIdx.x;
    if (e >= EN) return;
    int s, d;
    if (e < E) { s = ei[e]; d = ei[E + e]; } else { s = d = e - E; }
    float l = s_src[s] + s_dst[d];
    l = (l > 0.0f) ? l : NEG_SLOPE * l;
    logits[e] = l;
    atomicMax(mkey + d, f32_key(l));
}

// ---------------------------------------------------------------------------
// K4: one wave per edge: e = exp(logit - m[dst]); denom[dst] += e;
//     out[dst,:] += e * h[src,:]   (h and out are L2-resident: 25.6 MB each
//     vs 192 MB L2, so the random gathers/atomics never touch HBM)
//     evals overwrites logits in place.
// ---------------------------------------------------------------------------
__global__ __launch_bounds__(256) void gat_scatter(const int* __restrict__ ei,
                                                   const float* __restrict__ h,
                                                   float* __restrict__ evals,
                                                   const unsigned* __restrict__ mkey,
                                                   float* __restrict__ denom,
                                                   float* __restrict__ out,
                                                   int E, int EN) {
    int e    = (blockIdx.x * 256 + threadIdx.x) >> 5;
    int lane = threadIdx.x & 31;
    if (e >= EN) return;
    int s, d;
    if (e < E) { s = ei[e]; d = ei[E + e]; } else { s = d = e - E; }
    float m  = key_f32(mkey[d]);
    float l  = evals[e];               // all lanes load before lane0 overwrites
    float ev = expf(l - m);
    if (lane == 0) { evals[e] = ev; atomicAdd(denom + d, ev); }
    float2 hv = ((const float2*)(h + (size_t)s * OUT_CH))[lane];
    float* op = out + (size_t)d * OUT_CH + 2 * lane;
    atomicAdd(op,     ev * hv.x);
    atomicAdd(op + 1, ev * hv.y);
}

// ---------------------------------------------------------------------------
// K5: alpha[e] = e/denom[dst]; also emit edge_index_out (as exact f32 ints)
// ---------------------------------------------------------------------------
__global__ __launch_bounds__(256) void gat_alpha(const int* __restrict__ ei,
                                                 const float* __restrict__ evals,
                                                 const float* __restrict__ denom,
                                                 float* __restrict__ alpha_out,
                                                 float* __restrict__ eidx_out,
                                                 int E, int EN) {
    int e = blockIdx.x * 256 + threadIdx.x;
    if (e >= EN) return;
    int s, d;
    if (e < E) { s = ei[e]; d = ei[E + e]; } else { s = d = e - E; }
    alpha_out[e] = evals[e] / denom[d];
    eidx_out[e]      = (float)s;
    eidx_out[EN + e] = (float)d;
}

// ---------------------------------------------------------------------------
// K6: out = out/denom + bias
// ---------------------------------------------------------------------------
__global__ __launch_bounds__(256) void gat_finalize(float* __restrict__ out,
                                                    const float* __restrict__ denom,
                                                    const float* __restrict__ bias,
                                                    int total) {
    int i = blockIdx.x * 256 + threadIdx.x;
    if (i >= total) return;
    int n = i >> 6, c = i & 63;
    out[i] = out[i] / denom[n] + bias[c];
}

// ---------------------------------------------------------------------------
extern "C" void kernel_launch(void* const* d_in, const int* in_sizes, int n_in,
                              void* d_out, int out_size, void* d_ws, size_t ws_size,
                              hipStream_t stream) {
    (void)n_in; (void)out_size; (void)ws_size;

    const float* x     = (const float*)d_in[0];
    const int*   ei    = (const int*)d_in[1];
    const float* W     = (const float*)d_in[2];
    const float* a_src = (const float*)d_in[3];
    const float* a_dst = (const float*)d_in[4];
    const float* bias  = (const float*)d_in[5];

    const int N  = in_sizes[0] / IN_CH;   // 100000
    const int E  = in_sizes[1] / 2;       // 1600000
    const int EN = E + N;                 // 1700000

    // workspace carve-up
    char* ws = (char*)d_ws;
    float*    h      = (float*)ws;    ws += (size_t)N * OUT_CH * sizeof(float);
    float*    s_src  = (float*)ws;    ws += (size_t)N * sizeof(float);
    float*    s_dst  = (float*)ws;    ws += (size_t)N * sizeof(float);
    float*    denom  = (float*)ws;    ws += (size_t)N * sizeof(float);
    unsigned* mkey   = (unsigned*)ws; ws += (size_t)N * sizeof(unsigned);
    float*    evals  = (float*)ws;    ws += (size_t)EN * sizeof(float);

    // output carve-up (flat f32, tuple order: out, edge_index_out, alpha)
    float* out       = (float*)d_out;
    float* eidx_out  = out + (size_t)N * OUT_CH;
    float* alpha_out = eidx_out + (size_t)2 * EN;

    const int totOut = N * OUT_CH;

    { int g = (totOut > N ? totOut : N); g = (g + 255) / 256;
      gat_init<<<g, 256, 0, stream>>>(mkey, denom, out, N, totOut); }

    { int waves = (N + 15) / 16; int g = (waves + 7) / 8;
      gat_gemm_wmma<<<g, 256, 0, stream>>>(x, W, h, N); }

    { int g = (N + 7) / 8;  // one wave per node, 8 waves per block
      gat_scores<<<g, 256, 0, stream>>>(h, a_src, a_dst, s_src, s_dst, N); }

    { int g = (EN + 255) / 256;
      gat_logits<<<g, 256, 0, stream>>>(ei, s_src, s_dst, evals, mkey, E, EN); }

    { int g = (EN + 7) / 8; // one wave per edge
      gat_scatter<<<g, 256, 0, stream>>>(ei, h, evals, mkey, denom, out, E, EN); }

    { int g = (EN + 255) / 256;
      gat_alpha<<<g, 256, 0, stream>>>(ei, evals, denom, alpha_out, eidx_out, E, EN); }

    { int g = (totOut + 255) / 256;
      gat_finalize<<<g, 256, 0, stream>>>(out, denom, bias, totOut); }
}